// IntentSpace_51187420234135
// MI455X (gfx1250) — compile-verified
//
#include <hip/hip_runtime.h>
#include <hip/hip_bf16.h>
#include <math.h>

#define T_DIM 16384
#define D_DIM 1024
#define H_DIM 1024
#define B_DIM 8

#define NWG   64           // persistent workgroups for the recurrence
#define ROWS  16           // rows of u per workgroup (16*1024*4B = 64KB LDS)

typedef float v2f __attribute__((ext_vector_type(2)));
typedef float v8f __attribute__((ext_vector_type(8)));

// ---------------------------------------------------------------------------
// Kernel 0: init workspace (global barrier state + h0 = 0)
// ---------------------------------------------------------------------------
__global__ void init_ws(unsigned* __restrict__ bar, float* __restrict__ hbuf) {
  const int i = threadIdx.x;
  if (i < 2) bar[i] = 0u;
  if (i < H_DIM) hbuf[i] = 0.0f;   // h0 = 0 (ping buffer)
}

// ---------------------------------------------------------------------------
// Kernel 1: alpha = softmax(alphas[alpha_int]);  u = sum_b alpha_b * Ws_b
// ---------------------------------------------------------------------------
__global__ void compute_u(const float* __restrict__ alphas,
                          const int* __restrict__ alpha_int,
                          const float* __restrict__ Ws,
                          float* __restrict__ u) {
  const int i = blockIdx.x * blockDim.x + threadIdx.x;   // [0, H*H)
  const float* arow = alphas + (size_t)(*alpha_int) * B_DIM;
  float m = arow[0];
#pragma unroll
  for (int b = 1; b < B_DIM; ++b) m = fmaxf(m, arow[b]);
  float e[B_DIM];
  float s = 0.0f;
#pragma unroll
  for (int b = 0; b < B_DIM; ++b) { e[b] = __expf(arow[b] - m); s += e[b]; }
  const float inv = 1.0f / s;
  float acc = 0.0f;
#pragma unroll
  for (int b = 0; b < B_DIM; ++b)
    acc += (e[b] * inv) * Ws[(size_t)b * H_DIM * H_DIM + i];
  u[i] = acc;
}

// ---------------------------------------------------------------------------
// Kernel 2: ig = x @ v.T  via V_WMMA_F32_16X16X4_F32.
// One wave computes a 16x64 strip (4 N-tiles share one A-tile): per K-step
// 5 b64 loads feed 4 wmma ops (vs 8:4 for the naive 1-tile version).
//   ig[t][h] = sum_d x[t][d] * v[h][d]
// ---------------------------------------------------------------------------
__global__ __launch_bounds__(128)
void ig_gemm(const float* __restrict__ x,
             const float* __restrict__ v,
             float* __restrict__ ig) {
  const int lane    = threadIdx.x & 31;
  const int wave    = threadIdx.x >> 5;
  const int strip   = blockIdx.x * 4 + wave;      // (T/16)*(H/64) strips
  const int stripsN = H_DIM / 64;                 // 16
  const int tm = strip / stripsN;                 // 16-row tile index
  const int tn = strip % stripsN;                 // 64-col strip index
  const int l16   = lane & 15;
  const int khalf = lane >> 4;                    // 0: K{0,1}, 1: K{2,3}

  const float* arow = x + (size_t)(tm * 16 + l16) * D_DIM + khalf * 2;
  const float* b0 = v + (size_t)(tn * 64 +  0 + l16) * D_DIM + khalf * 2;
  const float* b1 = v + (size_t)(tn * 64 + 16 + l16) * D_DIM + khalf * 2;
  const float* b2 = v + (size_t)(tn * 64 + 32 + l16) * D_DIM + khalf * 2;
  const float* b3 = v + (size_t)(tn * 64 + 48 + l16) * D_DIM + khalf * 2;

  v8f acc0 = {}, acc1 = {}, acc2 = {}, acc3 = {};
#pragma unroll 2
  for (int k = 0; k < D_DIM; k += 4) {
    v2f a  = *(const v2f*)(arow + k);   // A[M][K..K+1]
    v2f q0 = *(const v2f*)(b0 + k);     // B[K..K+1][N] = v[N][K..K+1]
    v2f q1 = *(const v2f*)(b1 + k);
    v2f q2 = *(const v2f*)(b2 + k);
    v2f q3 = *(const v2f*)(b3 + k);
    acc0 = __builtin_amdgcn_wmma_f32_16x16x4_f32(false, a, false, q0,
                                                 (short)0, acc0, false, false);
    acc1 = __builtin_amdgcn_wmma_f32_16x16x4_f32(false, a, false, q1,
                                                 (short)0, acc1, false, false);
    acc2 = __builtin_amdgcn_wmma_f32_16x16x4_f32(false, a, false, q2,
                                                 (short)0, acc2, false, false);
    acc3 = __builtin_amdgcn_wmma_f32_16x16x4_f32(false, a, false, q3,
                                                 (short)0, acc3, false, false);
  }
  // C layout: VGPR j -> M = j + 8*(lane/16), N = lane%16
  const int mBase = tm * 16 + khalf * 8;
  float* og = ig + (size_t)mBase * H_DIM + tn * 64 + l16;
#pragma unroll
  for (int j = 0; j < 8; ++j) {
    og[(size_t)j * H_DIM +  0] = acc0[j];
    og[(size_t)j * H_DIM + 16] = acc1[j];
    og[(size_t)j * H_DIM + 32] = acc2[j];
    og[(size_t)j * H_DIM + 48] = acc3[j];
  }
}

// ---------------------------------------------------------------------------
// Device-wide sense barrier for the persistent recurrence kernel
// ---------------------------------------------------------------------------
__device__ __forceinline__ void grid_barrier(unsigned* cnt,
                                             volatile unsigned* sense,
                                             unsigned my_phase) {
  __syncthreads();
  if (threadIdx.x == 0) {
    __threadfence();
    if (atomicAdd(cnt, 1u) == NWG - 1) {
      *cnt = 0u;
      __threadfence();
      *sense = my_phase;
    } else {
      while (*sense < my_phase) __builtin_amdgcn_s_sleep(1);
    }
    __threadfence();
  }
  __syncthreads();
}

// ---------------------------------------------------------------------------
// Kernel 3: persistent recurrence. 64 WGs; WG g owns rows [16g, 16g+16) of u,
// kept in LDS (loaded once by the Tensor Data Mover). Each of the 16384 steps:
//   h_new[r] = tanh(ig[t][r] + dot(u[r], h) + b[r]);  device barrier per step.
// Tail: out[r] = sigmoid(dot(fc_w[r], h_final) + fc_b[r]).
// ---------------------------------------------------------------------------
__global__ __launch_bounds__(256, 1)
void rnn_scan(const float* __restrict__ u,
              const float* __restrict__ ig,
              const float* __restrict__ bias,
              const float* __restrict__ fc_w,
              const float* __restrict__ fc_b,
              float* __restrict__ hbuf,     // 2*H floats, ping-pong
              unsigned* __restrict__ bar,   // [cnt, sense]
              float* __restrict__ out) {
  __shared__ float uslice[ROWS * H_DIM];    // 64 KB slab of u
  __shared__ float hl[H_DIM];               // current h staged in LDS

  const int wg  = blockIdx.x;
  const int tid = threadIdx.x;
  const int rowInWg = tid >> 4;             // 0..15  (16 threads per row)
  const int seg     = tid & 15;             // 64-wide column segment
  const int row     = wg * ROWS + rowInWg;

  // ---- load u slice into LDS via the Tensor Data Mover (2D D# descriptor) --
#if __has_builtin(__builtin_amdgcn_tensor_load_to_lds)
  if (tid < 32) {   // single wave issues the TDM op (EXEC ignored by TDM)
    typedef unsigned int u32x4 __attribute__((ext_vector_type(4)));
    typedef int i32x4 __attribute__((ext_vector_type(4)));
    typedef int i32x8 __attribute__((ext_vector_type(8)));
    const unsigned long long ga =
        (unsigned long long)(uintptr_t)(u + (size_t)wg * ROWS * H_DIM);
    const unsigned int lds = (unsigned int)(uintptr_t)(&uslice[0]);
    u32x4 g0;
    g0[0] = 1u;                                          // count=1, user desc
    g0[1] = lds;                                         // lds_addr (bytes)
    g0[2] = (unsigned int)(ga & 0xFFFFFFFFu);            // global_addr[31:0]
    g0[3] = (unsigned int)((ga >> 32) & 0x01FFFFFFu)     // global_addr[56:32]
          | (2u << 30);                                  // type = 2 (image)
    i32x8 g1;
    g1[0] = (int)(2u << 16);          // wg_mask=0, data_size=2 (4B), no flags
    g1[1] = (int)((unsigned)H_DIM << 16);   // tensor_dim0[15:0] -> bits 63:48
    g1[2] = (int)((unsigned)H_DIM << 16);   // dim0 hi=0 | tensor_dim1 lo16
    g1[3] = (int)((unsigned)H_DIM << 16);   // dim1 hi=0 | tile_dim0 = 1024
    g1[4] = (int)ROWS;                      // tile_dim1 = 16, tile_dim2 = 0
    g1[5] = (int)H_DIM;                     // tensor_dim0_stride lo32 = 1024
    g1[6] = 0;                              // stride hi | dim1_stride lo
    g1[7] = 0;
    i32x4 gz = {};
#if defined(__clang_major__) && (__clang_major__ >= 23)
    i32x8 gz8 = {};
    __builtin_amdgcn_tensor_load_to_lds(g0, g1, gz, gz, gz8, 0);
#else
    __builtin_amdgcn_tensor_load_to_lds(g0, g1, gz, gz, 0);
#endif
    __builtin_amdgcn_s_wait_tensorcnt(0);
  }
#else
  for (int i = tid; i < ROWS * H_DIM; i += 256)
    uslice[i] = u[(size_t)wg * ROWS * H_DIM + i];
#endif
  __syncthreads();

  const float brow = (seg == 0) ? bias[row] : 0.0f;
  float* const h0 = hbuf;
  float* const h1 = hbuf + H_DIM;
  unsigned phase = 0;

  for (int t = 0; t < T_DIM; ++t) {
    const float* hin  = (t & 1) ? h1 : h0;
    float*       hout = (t & 1) ? h0 : h1;

    // Stage h into LDS (volatile: bypass stale WGP$ after the global barrier)
    {
      const volatile float* hv = hin;
      for (int i = tid; i < H_DIM; i += 256) hl[i] = hv[i];
    }
    __syncthreads();

    // dot(u[row], h) -- each thread covers a 64-wide segment out of LDS
    const float* ur = &uslice[rowInWg * H_DIM + seg * 64];
    const float* hs = &hl[seg * 64];
    float s = 0.0f;
#pragma unroll 8
    for (int c = 0; c < 64; ++c) s = fmaf(ur[c], hs[c], s);
    s += __shfl_xor(s, 8, 32);
    s += __shfl_xor(s, 4, 32);
    s += __shfl_xor(s, 2, 32);
    s += __shfl_xor(s, 1, 32);
    if (seg == 0)
      hout[row] = tanhf(ig[(size_t)t * H_DIM + row] + s + brow);
    __syncthreads();           // protect hl before next-step overwrite

    grid_barrier(&bar[0], &bar[1], ++phase);
  }

  // ---- tail: out = sigmoid(h @ fc_w.T + fc_b), final h is in h0 (T even) --
  {
    const volatile float* hv = h0;
    for (int i = tid; i < H_DIM; i += 256) hl[i] = hv[i];
  }
  __syncthreads();
  const float* fr = fc_w + (size_t)row * H_DIM + seg * 64;
  const float* hs = &hl[seg * 64];
  float s = 0.0f;
#pragma unroll 8
  for (int c = 0; c < 64; ++c) s = fmaf(fr[c], hs[c], s);
  s += __shfl_xor(s, 8, 32);
  s += __shfl_xor(s, 4, 32);
  s += __shfl_xor(s, 2, 32);
  s += __shfl_xor(s, 1, 32);
  if (seg == 0)
    out[row] = 1.0f / (1.0f + __expf(-(s + fc_b[row])));
}

// ---------------------------------------------------------------------------
extern "C" void kernel_launch(void* const* d_in, const int* in_sizes, int n_in,
                              void* d_out, int out_size, void* d_ws, size_t ws_size,
                              hipStream_t stream) {
  const float* x        = (const float*)d_in[0];   // (T, D)
  const float* alphas   = (const float*)d_in[1];   // (B, B)
  const float* Ws       = (const float*)d_in[2];   // (B, H, H)
  const float* bias     = (const float*)d_in[3];   // (H, 1)
  const float* v        = (const float*)d_in[4];   // (H, D)
  const float* fc_w     = (const float*)d_in[5];   // (H, H)
  const float* fc_b     = (const float*)d_in[6];   // (H,)
  const int*   alpha_i  = (const int*)d_in[7];     // scalar
  float* out = (float*)d_out;                      // (1, H)

  char* ws = (char*)d_ws;
  unsigned* bar  = (unsigned*)ws;                       // [0, 256)
  float*    hbuf = (float*)(ws + 256);                  // 2*H floats
  float*    u    = (float*)(ws + (16u << 10));          // 4 MB   @ 16 KB
  float*    ig   = (float*)(ws + (8u  << 20));          // 64 MB  @ 8 MB

  init_ws<<<1, 1024, 0, stream>>>(bar, hbuf);
  compute_u<<<(H_DIM * H_DIM) / 256, 256, 0, stream>>>(alphas, alpha_i, Ws, u);
  ig_gemm<<<(T_DIM / 16) * (H_DIM / 64) / 4, 128, 0, stream>>>(x, v, ig);
  rnn_scan<<<NWG, 256, 0, stream>>>(u, ig, bias, fc_w, fc_b, hbuf, bar, out);
}